// CellNet_10041633538768
// MI455X (gfx1250) — compile-verified
//
#include <hip/hip_runtime.h>

// CDNA5 / gfx1250, wave32.
typedef __attribute__((ext_vector_type(2))) float v2f;
typedef __attribute__((ext_vector_type(8))) float v8f;

// ---------------------------------------------------------------- utilities
__global__ void zero_f(float* __restrict__ p, int n) {
    int i = blockIdx.x * blockDim.x + threadIdx.x;
    if (i < n) p[i] = 0.f;
}

__global__ void deg_kernel(const int* __restrict__ dst, float* __restrict__ cnt, int E) {
    int e = blockIdx.x * blockDim.x + threadIdx.x;
    if (e < E) atomicAdd(&cnt[dst[e]], 1.0f);
}

// ---------------------------------------------------------------- edge kernel
// One wave handles 16 edges per group; block = 4 waves * EPB groups = 512 edges.
// Edge-NN linear terms via V_WMMA_F32_16X16X4_F32:
//   A = [ef0 ef1 1 0] (16 edges x 4), B tile t: columns c = t*OUT_C + n of the
//   combined 4xD table [Wh0; Wh1; bh; 0] (branch-free: lo lanes read rows 0,1,
//   hi lanes rows 2,3 via koff).
// Tile index t == input channel i, lane column n == output channel o, so each lane
// owns a fixed (rows, o) slice and accumulates messages in registers.
// A-frag (32-bit A 16x4): lanes 0-15 hold K=0,1 ; lanes 16-31 hold K=2,3.
// D-frag (32-bit C/D 16x16): VGPR j, lane l -> (M = j + 8*(l>=16), N = l&15).
constexpr int EPB = 8;  // 64-edge groups per block

template <int IN_C, int OUT_C>
__global__ __launch_bounds__(128) void edge_kernel(
    const float* __restrict__ x, const float* __restrict__ ef,
    const int* __restrict__ et, const int* __restrict__ esrc, const int* __restrict__ edst,
    const float* __restrict__ emb, const float* __restrict__ wh, const float* __restrict__ bh,
    const float* __restrict__ wg, const float* __restrict__ bg,
    float* __restrict__ s, int E) {
    constexpr int D = IN_C * OUT_C;

    __shared__ float emb_s[25 * D];
    __shared__ float bth[4 * D];    // rows: Wh0, Wh1, bh, 0
    __shared__ float btg[4 * D];    // rows: Wg0, Wg1, bg, 0
    __shared__ int   met_etD[64];   // edge_type * D
    __shared__ int   met_dst[64];
    __shared__ float xs[64][IN_C];  // gathered x[src]

    const int tid = threadIdx.x;    // 128 = 4 waves

    // Stage tables into LDS once per block.
    for (int i = tid; i < 25 * D; i += 128) emb_s[i] = emb[i];
    for (int i = tid; i < D; i += 128) {
        bth[i] = wh[i];  bth[D + i] = wh[D + i];  bth[2 * D + i] = bh[i];  bth[3 * D + i] = 0.f;
        btg[i] = wg[i];  btg[D + i] = wg[D + i];  btg[2 * D + i] = bg[i];  btg[3 * D + i] = 0.f;
    }

    const int  wid  = tid >> 5;
    const int  lane = tid & 31;
    const bool hi   = lane >= 16;
    const int  nlo  = lane & 15;            // == output channel o for this lane
    const int  lw   = wid * 16;
    const int  koff = hi ? 2 * D : 0;       // which pair of K-rows this lane supplies

    for (int g = 0; g < EPB; ++g) {
        const int gbase = (blockIdx.x * EPB + g) * 64;
        __syncthreads();            // protect xs/meta (and tables on first pass)
        if (tid < 64) {
            int e = gbase + tid;
            bool v = e < E;
            int srcn = v ? esrc[e] : 0;
            met_etD[tid] = (v ? et[e] : 0) * D;
            met_dst[tid] = v ? edst[e] : 0;
            #pragma unroll
            for (int i = 0; i < IN_C; ++i)
                xs[tid][i] = v ? x[(size_t)srcn * IN_C + i] : 0.f;
        }
        // Speculative prefetch of next group's edge features (global_prefetch_b8).
        __builtin_prefetch(ef + 2 * (size_t)(gbase + 64) + 2 * (size_t)tid, 0, 1);
        __syncthreads();

        // A fragment (bias folded via constant-1 K-slot), built once per group.
        v2f a;
        if (!hi) {
            int e = gbase + lw + nlo;
            bool v = e < E;
            a.x = v ? ef[2 * (size_t)e]     : 0.f;   // K=0: ef0
            a.y = v ? ef[2 * (size_t)e + 1] : 0.f;   // K=1: ef1
        } else {
            a.x = 1.f;                               // K=2: bias
            a.y = 0.f;                               // K=3: pad
        }

        // Hoist per-row metadata for this lane's 8 rows.
        int rowv[8], ebD[8];
        #pragma unroll
        for (int j = 0; j < 8; ++j) {
            rowv[j] = lw + j + (hi ? 8 : 0);
            ebD[j]  = met_etD[rowv[j]];
        }

        float acc[8];
        #pragma unroll
        for (int j = 0; j < 8; ++j) acc[j] = 0.f;

        #pragma unroll 2                           // cap register pressure
        for (int t = 0; t < IN_C; ++t) {           // tile == input channel i
            int c  = t * OUT_C + nlo;              // d-column for this lane
            int cc = c < D ? c : (D - 1);          // clamp (cols >= OUT_C unused)
            // Branch-free B fragments: every lane loads its two K-rows.
            v2f bhf, bgf;
            bhf.x = bth[koff + cc];  bhf.y = bth[koff + D + cc];
            bgf.x = btg[koff + cc];  bgf.y = btg[koff + D + cc];
            v8f cz = {};
            v8f H = __builtin_amdgcn_wmma_f32_16x16x4_f32(false, a, false, bhf, (short)0, cz, false, false);
            v8f G = __builtin_amdgcn_wmma_f32_16x16x4_f32(false, a, false, bgf, (short)0, cz, false, false);
            #pragma unroll
            for (int j = 0; j < 8; ++j) {          // branch-free elementwise tail
                float w = emb_s[ebD[j] + cc] * H[j] + G[j];
                w = w > 0.f ? w : 0.f;             // relu -> per-edge weight
                acc[j] += xs[rowv[j]][t] * w;      // msg accumulation in registers
            }
        }

        // Flush: each (edge,o) owned by exactly one lane -> one global atomic each.
        if (nlo < OUT_C) {
            #pragma unroll
            for (int j = 0; j < 8; ++j) {
                int e = gbase + rowv[j];
                if (e < E)
                    atomicAdd(&s[(size_t)met_dst[rowv[j]] * OUT_C + nlo], acc[j]);
            }
        }
    }
}

// ---------------------------------------------------------------- node update
// h_out = relu( s/max(cnt,1) + x@root + bias ), x@root via WMMA f32 16x16x4 K-loop.
template <int IN_C, int OUT_C>
__global__ __launch_bounds__(128) void node_kernel(
    const float* __restrict__ x, const float* __restrict__ root, const float* __restrict__ bias,
    const float* __restrict__ s, const float* __restrict__ cnt,
    float* __restrict__ hout, int N) {
    constexpr int KT = (IN_C + 3) / 4;
    const int tid = threadIdx.x, wid = tid >> 5, lane = tid & 31;
    const int r0  = blockIdx.x * 64 + wid * 16;   // 16-node M-tile per wave
    const bool hi = lane >= 16;
    const int  n  = lane & 15;

    v8f acc = {};
    for (int kt = 0; kt < KT; ++kt) {
        int k0 = kt * 4;
        int ka = k0 + (hi ? 2 : 0);
        int m  = lane & 15;
        int row = r0 + m;
        bool rv = row < N;
        v2f a, b;
        a.x = (rv && ka     < IN_C) ? x[(size_t)row * IN_C + ka]     : 0.f;
        a.y = (rv && ka + 1 < IN_C) ? x[(size_t)row * IN_C + ka + 1] : 0.f;
        b.x = (ka     < IN_C && n < OUT_C) ? root[(size_t)ka       * OUT_C + n] : 0.f;
        b.y = (ka + 1 < IN_C && n < OUT_C) ? root[(size_t)(ka + 1) * OUT_C + n] : 0.f;
        acc = __builtin_amdgcn_wmma_f32_16x16x4_f32(false, a, false, b, (short)0, acc, false, false);
    }

    if (n < OUT_C) {
        #pragma unroll
        for (int j = 0; j < 8; ++j) {
            int row = r0 + j + (hi ? 8 : 0);
            if (row < N) {
                float cn = cnt[row]; cn = cn > 1.f ? cn : 1.f;
                float v = s[(size_t)row * OUT_C + n] / cn + acc[j] + bias[n];
                hout[(size_t)row * OUT_C + n] = v > 0.f ? v : 0.f;
            }
        }
    }
}

// ---------------------------------------------------------------- pooling
__global__ void pool_accum(const float* __restrict__ h, const int* __restrict__ cell_type,
                           const int* __restrict__ batch_ids,
                           float* __restrict__ psum, float* __restrict__ pcnt, int N) {
    int k = blockIdx.x * blockDim.x + threadIdx.x;
    int node = k >> 4, o = k & 15;
    if (node < N && cell_type[node] == 1) {
        int b = batch_ids[node] & 63;
        atomicAdd(&psum[b * 16 + o], h[(size_t)node * 16 + o]);
        if (o == 0) atomicAdd(&pcnt[b], 1.0f);
    }
}

__global__ void pool_final(const float* __restrict__ psum, const float* __restrict__ pcnt,
                           float* __restrict__ out, int out_size) {
    int t = blockIdx.x * blockDim.x + threadIdx.x;
    if (t < out_size) {
        int b = t >> 4;
        float c = pcnt[b]; c = c > 1.f ? c : 1.f;
        out[t] = psum[t] / c;
    }
}

// ---------------------------------------------------------------- launch
extern "C" void kernel_launch(void* const* d_in, const int* in_sizes, int n_in,
                              void* d_out, int out_size, void* d_ws, size_t ws_size,
                              hipStream_t stream) {
    (void)n_in; (void)ws_size;
    const float* x    = (const float*)d_in[0];
    const float* ef   = (const float*)d_in[1];
    const int*   et   = (const int*)d_in[2];
    const int*   esrc = (const int*)d_in[3];
    const int*   edst = (const int*)d_in[4];
    const int*   ctyp = (const int*)d_in[5];
    const int*   bid  = (const int*)d_in[6];
    // d_in[7] = batch_size (device scalar; encoded in out_size anyway)
    const float* P[21];
    for (int i = 0; i < 21; ++i) P[i] = (const float*)d_in[8 + i];
    const float** L1 = &P[0];  const float** L2 = &P[7];  const float** L3 = &P[14];

    const int N = in_sizes[0] / 16;   // x is [N,16]
    const int E = in_sizes[2];        // edge_type count

    float* ws   = (float*)d_ws;
    float* cnt  = ws;                 // [N]
    float* s    = cnt  + N;           // [N,16] segment sums (reused per layer)
    float* h1   = s    + (size_t)N * 16;
    float* h2   = h1   + (size_t)N * 16;
    float* h3   = h2   + (size_t)N * 16;
    float* psum = h3   + (size_t)N * 16;  // [64,16]
    float* pcnt = psum + 1024;            // [64]

    const int ZT = 256;
    zero_f<<<(N + ZT - 1) / ZT, ZT, 0, stream>>>(cnt, N);
    zero_f<<<(1088 + ZT - 1) / ZT, ZT, 0, stream>>>(psum, 1088);   // psum + pcnt
    deg_kernel<<<(E + ZT - 1) / ZT, ZT, 0, stream>>>(edst, cnt, E);

    const int EB = (E + 64 * EPB - 1) / (64 * EPB);
    const int NB = (N + 63) / 64;

    // Layer 1: 16 -> 10
    zero_f<<<(N * 10 + ZT - 1) / ZT, ZT, 0, stream>>>(s, N * 10);
    edge_kernel<16, 10><<<EB, 128, 0, stream>>>(x, ef, et, esrc, edst,
        L1[0], L1[1], L1[2], L1[3], L1[4], s, E);
    node_kernel<16, 10><<<NB, 128, 0, stream>>>(x, L1[5], L1[6], s, cnt, h1, N);

    // Layer 2: 10 -> 10
    zero_f<<<(N * 10 + ZT - 1) / ZT, ZT, 0, stream>>>(s, N * 10);
    edge_kernel<10, 10><<<EB, 128, 0, stream>>>(h1, ef, et, esrc, edst,
        L2[0], L2[1], L2[2], L2[3], L2[4], s, E);
    node_kernel<10, 10><<<NB, 128, 0, stream>>>(h1, L2[5], L2[6], s, cnt, h2, N);

    // Layer 3: 10 -> 16
    zero_f<<<(N * 16 + ZT - 1) / ZT, ZT, 0, stream>>>(s, N * 16);
    edge_kernel<10, 16><<<EB, 128, 0, stream>>>(h2, ef, et, esrc, edst,
        L3[0], L3[1], L3[2], L3[3], L3[4], s, E);
    node_kernel<10, 16><<<NB, 128, 0, stream>>>(h2, L3[5], L3[6], s, cnt, h3, N);

    // Gated scatter-mean pooling
    pool_accum<<<(N * 16 + ZT - 1) / ZT, ZT, 0, stream>>>(h3, ctyp, bid, psum, pcnt, N);
    pool_final<<<1, 256, 0, stream>>>(psum, pcnt, (float*)d_out, out_size);
}